// GCN_30021821399331
// MI455X (gfx1250) — compile-verified
//
#include <hip/hip_runtime.h>

typedef __attribute__((ext_vector_type(2))) float v2f;
typedef __attribute__((ext_vector_type(8))) float v8f;

#define LRELU(v) ((v) > 0.0f ? (v) : 0.2f * (v))
#define GAT_EPS 1e-16f
#define NEG_INF (-__builtin_inff())

#if __has_builtin(__builtin_amdgcn_global_load_async_to_lds_b128)
#define HAVE_ASYNC_LDS 1
#endif

#ifdef HAVE_ASYNC_LDS
typedef __attribute__((__vector_size__(4 * sizeof(int)))) int v4i_vs;
#define AS_GLOBAL(p) ((__attribute__((address_space(1))) v4i_vs*)(p))
#define AS_LOCAL(p)  ((__attribute__((address_space(3))) v4i_vs*)(p))
__device__ __forceinline__ void wait_asynccnt0() {
#if __has_builtin(__builtin_amdgcn_s_wait_asynccnt)
  __builtin_amdgcn_s_wait_asynccnt(0);
#else
  asm volatile("s_wait_asynccnt 0x0" ::: "memory");
#endif
}
#endif

// -------- generic fill --------
__global__ void fill_kernel(float* p, float val, int n) {
  int i = blockIdx.x * blockDim.x + threadIdx.x;
  if (i < n) p[i] = val;
}

// -------- float atomic max via sign-split integer atomics --------
__device__ __forceinline__ void atomicMaxFloat(float* addr, float value) {
  if (value >= 0.0f)
    atomicMax((int*)addr, __float_as_int(value));
  else
    atomicMin((unsigned int*)addr, __float_as_uint(value));
}

// -------- Layer-1 GEMM: h1[N,32] = x[N,128] @ W1[128,32] via fp32 WMMA --------
// 128 threads = 4 waves; each wave computes a 16x32 tile; 64 rows per block.
// Tiles staged into LDS with the CDNA5 async global->LDS path (ASYNCcnt).
__global__ __launch_bounds__(128) void gemm1_wmma(
    const float* __restrict__ x, const float* __restrict__ W1,
    float* __restrict__ h1, int N) {
  __shared__ float sx[64 * 132];   // 64 rows x 128 cols, stride 132 (bank-safe, 16B-aligned rows)
  __shared__ float sw[128 * 40];   // 128 rows x 32 cols, stride 40 (bank-safe, 16B-aligned rows)
  const int t = threadIdx.x;
  const int rowBase = blockIdx.x * 64;

  // ---- stage W1 (128x32 = 1024 float4, none cross a padded row boundary) ----
  for (int i = t; i < 1024; i += 128) {
    int r = i >> 3;
    int c = (i & 7) << 2;
#ifdef HAVE_ASYNC_LDS
    __builtin_amdgcn_global_load_async_to_lds_b128(
        AS_GLOBAL(W1 + (i << 2)), AS_LOCAL(sw + r * 40 + c), 0, 0);
#else
    float4 v = *(const float4*)(W1 + (i << 2));
    float* dst = &sw[r * 40 + c];
    dst[0] = v.x; dst[1] = v.y; dst[2] = v.z; dst[3] = v.w;
#endif
  }
  // ---- stage x tile (64x128 = 2048 float4) ----
  for (int i = t; i < 2048; i += 128) {
    int r = i >> 5;
    int c = (i & 31) << 2;
    int gr = rowBase + r;
    float* dst = &sx[r * 132 + c];
    if (gr < N) {
#ifdef HAVE_ASYNC_LDS
      __builtin_amdgcn_global_load_async_to_lds_b128(
          AS_GLOBAL(x + (size_t)gr * 128 + c), AS_LOCAL(dst), 0, 0);
#else
      float4 v = *(const float4*)(x + (size_t)gr * 128 + c);
      dst[0] = v.x; dst[1] = v.y; dst[2] = v.z; dst[3] = v.w;
#endif
    } else {
      dst[0] = 0.f; dst[1] = 0.f; dst[2] = 0.f; dst[3] = 0.f;
    }
  }
#ifdef HAVE_ASYNC_LDS
  wait_asynccnt0();   // own async transfers landed in LDS
#endif
  __syncthreads();    // all waves' transfers visible

  const int wave = t >> 5;      // wave32
  const int lane = t & 31;
  const int lm   = lane & 15;   // lane % 16
  const int lg   = lane >> 4;   // half-wave select
  const int mrow = wave * 16;   // tile-local row base

  v8f c0 = {}; v8f c1 = {};
  #pragma unroll
  for (int kk = 0; kk < 32; ++kk) {
    const int k = kk * 4;
    v2f a, b0, b1;
    // A frag: M = lm, K = (2*lg + r)
    const float* ap = &sx[(mrow + lm) * 132 + k + 2 * lg];
    a.x = ap[0]; a.y = ap[1];
    // B frag: K = (2*lg + r), Ncol = lm (+16 for second column tile)
    const float* bp = &sw[(k + 2 * lg) * 40 + lm];
    b0.x = bp[0];  b0.y = bp[40];
    b1.x = bp[16]; b1.y = bp[56];
    c0 = __builtin_amdgcn_wmma_f32_16x16x4_f32(false, a, false, b0, (short)0, c0, false, false);
    c1 = __builtin_amdgcn_wmma_f32_16x16x4_f32(false, a, false, b1, (short)0, c1, false, false);
  }
  #pragma unroll
  for (int v = 0; v < 8; ++v) {
    int gr = rowBase + mrow + v + 8 * lg;   // C/D frag: M = v + 8*lg, N = lm
    if (gr < N) {
      h1[(size_t)gr * 32 + lm]      = c0[v];
      h1[(size_t)gr * 32 + 16 + lm] = c1[v];
    }
  }
}

// -------- attention logits, layer 1: as1/ad1[N,4] --------
__global__ void attn1_kernel(const float* __restrict__ h1,
                             const float* __restrict__ Asrc,
                             const float* __restrict__ Adst,
                             float* __restrict__ as1, float* __restrict__ ad1, int N) {
  int tid = blockIdx.x * blockDim.x + threadIdx.x;
  if (tid >= N * 4) return;
  int n = tid >> 2, h = tid & 3;
  const float* hp = h1 + (size_t)n * 32 + h * 8;
  float s = 0.f, d = 0.f;
  #pragma unroll
  for (int c = 0; c < 8; ++c) {
    float hv = hp[c];
    s += hv * Asrc[h * 8 + c];
    d += hv * Adst[h * 8 + c];
  }
  as1[tid] = s;
  ad1[tid] = d;
}

__device__ __forceinline__ void edge_ends(const long long* __restrict__ ei,
                                          int e, int E, int& s, int& d) {
  if (e < E) { s = (int)ei[e]; d = (int)ei[E + e]; }
  else       { s = e - E; d = s; }
}

// -------- layer 1 edge passes (4 heads) --------
__global__ void edge_max1(const long long* __restrict__ ei,
                          const float* __restrict__ as1, const float* __restrict__ ad1,
                          float* __restrict__ m1, int E, int ET) {
  int e = blockIdx.x * blockDim.x + threadIdx.x;
  if (e >= ET) return;
  int s, d; edge_ends(ei, e, E, s, d);
  #pragma unroll
  for (int h = 0; h < 4; ++h) {
    float v = LRELU(as1[s * 4 + h] + ad1[d * 4 + h]);
    atomicMaxFloat(&m1[d * 4 + h], v);
  }
}

__global__ void edge_sum1(const long long* __restrict__ ei,
                          const float* __restrict__ as1, const float* __restrict__ ad1,
                          const float* __restrict__ m1, float* __restrict__ s1,
                          int E, int ET) {
  int e = blockIdx.x * blockDim.x + threadIdx.x;
  if (e >= ET) return;
  int s, d; edge_ends(ei, e, E, s, d);
  #pragma unroll
  for (int h = 0; h < 4; ++h) {
    float v = LRELU(as1[s * 4 + h] + ad1[d * 4 + h]);
    atomicAdd(&s1[d * 4 + h], __expf(v - m1[d * 4 + h]));
  }
}

__global__ void edge_agg1(const long long* __restrict__ ei,
                          const float* __restrict__ as1, const float* __restrict__ ad1,
                          const float* __restrict__ m1, const float* __restrict__ s1,
                          const float* __restrict__ h1, float* __restrict__ agg,
                          int E, int ET) {
  int tid = blockIdx.x * blockDim.x + threadIdx.x;
  if (tid >= ET * 4) return;
  int e = tid >> 2, h = tid & 3;
  int s, d; edge_ends(ei, e, E, s, d);
  float v = LRELU(as1[s * 4 + h] + ad1[d * 4 + h]);
  float w = __expf(v - m1[d * 4 + h]);
  float alpha = w / (s1[d * 4 + h] + GAT_EPS);
  const float* hp = h1 + (size_t)s * 32 + h * 8;
  float* op = agg + (size_t)d * 32 + h * 8;
  #pragma unroll
  for (int c = 0; c < 8; ++c) atomicAdd(&op[c], hp[c] * alpha);
}

// -------- bias + ELU (in place) --------
__global__ void elu_bias1(float* __restrict__ agg, const float* __restrict__ b1, int n) {
  int i = blockIdx.x * blockDim.x + threadIdx.x;
  if (i >= n) return;
  float v = agg[i] + b1[i & 31];
  agg[i] = v > 0.f ? v : expm1f(v);
}

// -------- Layer-2 GEMM (F=32 -> 2) + attention logits; VALU (WMMA would waste 7/8 lanes) ----
__global__ void gemm2_kernel(const float* __restrict__ hin, const float* __restrict__ W2,
                             const float* __restrict__ As2, const float* __restrict__ Ad2,
                             float* __restrict__ h2, float* __restrict__ as2,
                             float* __restrict__ ad2, int N) {
  int n = blockIdx.x * blockDim.x + threadIdx.x;
  if (n >= N) return;
  const float* hp = hin + (size_t)n * 32;
  float a0 = 0.f, a1 = 0.f;
  #pragma unroll
  for (int k = 0; k < 32; ++k) {
    float hv = hp[k];
    a0 += hv * W2[2 * k];
    a1 += hv * W2[2 * k + 1];
  }
  h2[n * 2] = a0; h2[n * 2 + 1] = a1;
  as2[n] = a0 * As2[0] + a1 * As2[1];
  ad2[n] = a0 * Ad2[0] + a1 * Ad2[1];
}

// -------- layer 2 edge passes (1 head, 2 channels) --------
__global__ void edge_max2(const long long* __restrict__ ei,
                          const float* __restrict__ as2, const float* __restrict__ ad2,
                          float* __restrict__ m2, int E, int ET) {
  int e = blockIdx.x * blockDim.x + threadIdx.x;
  if (e >= ET) return;
  int s, d; edge_ends(ei, e, E, s, d);
  atomicMaxFloat(&m2[d], LRELU(as2[s] + ad2[d]));
}

__global__ void edge_sum2(const long long* __restrict__ ei,
                          const float* __restrict__ as2, const float* __restrict__ ad2,
                          const float* __restrict__ m2, float* __restrict__ s2,
                          int E, int ET) {
  int e = blockIdx.x * blockDim.x + threadIdx.x;
  if (e >= ET) return;
  int s, d; edge_ends(ei, e, E, s, d);
  float v = LRELU(as2[s] + ad2[d]);
  atomicAdd(&s2[d], __expf(v - m2[d]));
}

__global__ void edge_agg2(const long long* __restrict__ ei,
                          const float* __restrict__ as2, const float* __restrict__ ad2,
                          const float* __restrict__ m2, const float* __restrict__ s2,
                          const float* __restrict__ h2, float* __restrict__ out,
                          int E, int ET) {
  int e = blockIdx.x * blockDim.x + threadIdx.x;
  if (e >= ET) return;
  int s, d; edge_ends(ei, e, E, s, d);
  float v = LRELU(as2[s] + ad2[d]);
  float w = __expf(v - m2[d]);
  float alpha = w / (s2[d] + GAT_EPS);
  atomicAdd(&out[d * 2],     h2[s * 2]     * alpha);
  atomicAdd(&out[d * 2 + 1], h2[s * 2 + 1] * alpha);
}

__global__ void bias2_kernel(float* __restrict__ out, const float* __restrict__ b2, int n) {
  int i = blockIdx.x * blockDim.x + threadIdx.x;
  if (i < n) out[i] += b2[i & 1];
}

extern "C" void kernel_launch(void* const* d_in, const int* in_sizes, int n_in,
                              void* d_out, int out_size, void* d_ws, size_t ws_size,
                              hipStream_t stream) {
  (void)n_in; (void)out_size; (void)ws_size;
  const float*     x   = (const float*)d_in[0];
  const long long* ei  = (const long long*)d_in[1];
  const float*     W1  = (const float*)d_in[2];
  const float*     As1 = (const float*)d_in[3];
  const float*     Ad1 = (const float*)d_in[4];
  const float*     b1  = (const float*)d_in[5];
  const float*     W2  = (const float*)d_in[6];
  const float*     As2 = (const float*)d_in[7];
  const float*     Ad2 = (const float*)d_in[8];
  const float*     b2  = (const float*)d_in[9];
  float* out = (float*)d_out;

  const int N  = in_sizes[0] / 128;
  const int E  = in_sizes[1] / 2;
  const int ET = E + N;

  float* ws  = (float*)d_ws;
  float* h1   = ws;  ws += (size_t)N * 32;
  float* agg1 = ws;  ws += (size_t)N * 32;   // becomes elu(agg1+b1) in place
  float* as1  = ws;  ws += (size_t)N * 4;
  float* ad1  = ws;  ws += (size_t)N * 4;
  float* m1   = ws;  ws += (size_t)N * 4;
  float* s1   = ws;  ws += (size_t)N * 4;
  float* h2   = ws;  ws += (size_t)N * 2;
  float* as2  = ws;  ws += (size_t)N;
  float* ad2  = ws;  ws += (size_t)N;
  float* m2   = ws;  ws += (size_t)N;
  float* s2   = ws;  ws += (size_t)N;

  const int B = 256;
  auto nb = [](long long n, int b) { return (int)((n + b - 1) / b); };

  // ---- layer 1 ----
  gemm1_wmma<<<nb(N, 64), 128, 0, stream>>>(x, W1, h1, N);
  attn1_kernel<<<nb((long long)N * 4, B), B, 0, stream>>>(h1, As1, Ad1, as1, ad1, N);
  fill_kernel<<<nb((long long)N * 4, B), B, 0, stream>>>(m1, NEG_INF, N * 4);
  fill_kernel<<<nb((long long)N * 4, B), B, 0, stream>>>(s1, 0.f, N * 4);
  fill_kernel<<<nb((long long)N * 32, B), B, 0, stream>>>(agg1, 0.f, N * 32);
  edge_max1<<<nb(ET, B), B, 0, stream>>>(ei, as1, ad1, m1, E, ET);
  edge_sum1<<<nb(ET, B), B, 0, stream>>>(ei, as1, ad1, m1, s1, E, ET);
  edge_agg1<<<nb((long long)ET * 4, B), B, 0, stream>>>(ei, as1, ad1, m1, s1, h1, agg1, E, ET);
  elu_bias1<<<nb((long long)N * 32, B), B, 0, stream>>>(agg1, b1, N * 32);

  // ---- layer 2 ----
  gemm2_kernel<<<nb(N, B), B, 0, stream>>>(agg1, W2, As2, Ad2, h2, as2, ad2, N);
  fill_kernel<<<nb(N, B), B, 0, stream>>>(m2, NEG_INF, N);
  fill_kernel<<<nb(N, B), B, 0, stream>>>(s2, 0.f, N);
  fill_kernel<<<nb((long long)N * 2, B), B, 0, stream>>>(out, 0.f, N * 2);
  edge_max2<<<nb(ET, B), B, 0, stream>>>(ei, as2, ad2, m2, E, ET);
  edge_sum2<<<nb(ET, B), B, 0, stream>>>(ei, as2, ad2, m2, s2, E, ET);
  edge_agg2<<<nb(ET, B), B, 0, stream>>>(ei, as2, ad2, m2, s2, h2, out, E, ET);
  bias2_kernel<<<nb((long long)N * 2, B), B, 0, stream>>>(out, b2, N * 2);
}